// _SpaPayment_88399016886488
// MI455X (gfx1250) — compile-verified
//
#include <hip/hip_runtime.h>

// Second-price-auction "leave-one-out max" payment kernel for MI455X (gfx1250).
//
// Pure streaming workload: 256 MB in + 256 MB out, no reuse -> HBM-bound
// (~22 us floor at 23.3 TB/s). Strategy: 128-bit nontemporal loads/stores
// (global_load_b128 / global_store_b128 with th:*_NT so the 512 MB stream
// does not thrash the 192 MB L2), one 64-byte row per wave32 lane, branchless
// unrolled top-2 + argmax in registers.

typedef __attribute__((ext_vector_type(4))) float v4f;

__global__ __launch_bounds__(256) void spa_payment_kernel(
    const float* __restrict__ x, float* __restrict__ out, int nrows)
{
    int row = blockIdx.x * blockDim.x + threadIdx.x;
    if (row >= nrows) return;

    const v4f* in4 = reinterpret_cast<const v4f*>(x) + (size_t)row * 4;

    // 4x global_load_b128 th:TH_LOAD_NT (streamed once, don't cache)
    v4f a0 = __builtin_nontemporal_load(in4 + 0);
    v4f a1 = __builtin_nontemporal_load(in4 + 1);
    v4f a2 = __builtin_nontemporal_load(in4 + 2);
    v4f a3 = __builtin_nontemporal_load(in4 + 3);

    float v[16];
    v[ 0] = a0.x; v[ 1] = a0.y; v[ 2] = a0.z; v[ 3] = a0.w;
    v[ 4] = a1.x; v[ 5] = a1.y; v[ 6] = a1.z; v[ 7] = a1.w;
    v[ 8] = a2.x; v[ 9] = a2.y; v[10] = a2.z; v[11] = a2.w;
    v[12] = a3.x; v[13] = a3.y; v[14] = a3.z; v[15] = a3.w;

    // Branchless top-2 + argmax scan.
    // Strict '>' keeps the lowest index on ties (jax.lax.top_k semantics);
    // a duplicated max naturally yields max2 == max1 via the else-branch fmax.
    float m1 = v[0];
    float m2 = -__builtin_inff();
    int   arg = 0;
#pragma unroll
    for (int j = 1; j < 16; ++j) {
        float vj = v[j];
        bool  gt = vj > m1;                  // v_cmp_gt_f32
        m2  = gt ? m1 : fmaxf(m2, vj);       // v_max_num_f32 + v_cndmask
        arg = gt ? j  : arg;                 // v_cndmask
        m1  = gt ? vj : m1;                  // v_cndmask (== fmax under gt)
    }

    // Clamp at zero once; then select per column.
    float c1 = fmaxf(m1, 0.0f);
    float c2 = fmaxf(m2, 0.0f);

    float o[16];
#pragma unroll
    for (int j = 0; j < 16; ++j) {
        o[j] = (j == arg) ? c2 : c1;
    }

    v4f r0 = { o[ 0], o[ 1], o[ 2], o[ 3] };
    v4f r1 = { o[ 4], o[ 5], o[ 6], o[ 7] };
    v4f r2 = { o[ 8], o[ 9], o[10], o[11] };
    v4f r3 = { o[12], o[13], o[14], o[15] };

    v4f* out4 = reinterpret_cast<v4f*>(out) + (size_t)row * 4;

    // 4x global_store_b128 th:TH_STORE_NT
    __builtin_nontemporal_store(r0, out4 + 0);
    __builtin_nontemporal_store(r1, out4 + 1);
    __builtin_nontemporal_store(r2, out4 + 2);
    __builtin_nontemporal_store(r3, out4 + 3);
}

extern "C" void kernel_launch(void* const* d_in, const int* in_sizes, int n_in,
                              void* d_out, int out_size, void* d_ws, size_t ws_size,
                              hipStream_t stream)
{
    const float* x   = (const float*)d_in[0];
    float*       out = (float*)d_out;

    const int BIDDERS = 16;
    int nrows = in_sizes[0] / BIDDERS;   // 4194304

    const int threads = 256;             // 8 wave32s per block
    int blocks = (nrows + threads - 1) / threads;

    spa_payment_kernel<<<blocks, threads, 0, stream>>>(x, out, nrows);
}